// Protein_feature_58136677319451
// MI455X (gfx1250) — compile-verified
//
#include <hip/hip_runtime.h>
#include <math.h>

typedef __attribute__((ext_vector_type(2))) float v2f;
typedef __attribute__((ext_vector_type(8))) float v8f;

#define N_NODES 4096
#define TOPK    30
#define EPS_D   1e-6f

// ---------------- Kernel A: WMMA distance tiles + per-row top-30 ----------------
// Block = 128 threads = 4 waves. Each wave owns 32 rows (two 16-row WMMA tiles).
// Selection runs on SQUARED distances (monotone under sqrt); the hot loop has no
// sqrt and reads candidates back as float4 with a min4 pre-test so the branchy
// insertion path runs only on actual hits.

constexpr int WAVES_PER_BLOCK = 4;
constexpr int ROWS_PER_BLOCK  = 128;   // 4 waves * 32 rows
constexpr int TILE_LD         = 20;    // padded row stride: 80B, 16B-aligned

__global__ __launch_bounds__(128)
void knn_feat_kernel(const float* __restrict__ coords,
                     float* __restrict__ out_pos,   // (b,n,k,16)
                     float* __restrict__ out_gs,    // (b,n,k,15)
                     float* __restrict__ out_D,     // (b,n,k)
                     float* __restrict__ out_E)     // (b,n,k) as float
{
    __shared__ float sx[N_NODES];
    __shared__ float sy[N_NODES];
    __shared__ float sz[N_NODES];
    __shared__ __align__(16) float tile[WAVES_PER_BLOCK][32 * TILE_LD];

    const int nblk = N_NODES / ROWS_PER_BLOCK;           // 32
    const int b    = blockIdx.x / nblk;
    const int rb   = blockIdx.x % nblk;
    const float* Xb = coords + (size_t)b * N_NODES * 3;

    // Prefetch + stage coords SoA into LDS (uniform loop, no divergence)
    __builtin_prefetch(Xb + threadIdx.x * 16, 0, 0);     // global_prefetch_b8
    for (int t = threadIdx.x; t < N_NODES; t += ROWS_PER_BLOCK) {
        sx[t] = Xb[t * 3 + 0];
        sy[t] = Xb[t * 3 + 1];
        sz[t] = Xb[t * 3 + 2];
    }
    __syncthreads();

    const int wave = threadIdx.x >> 5;
    const int lane = threadIdx.x & 31;
    const int lm   = lane & 15;
    const bool hi  = lane >= 16;
    const int r0   = rb * ROWS_PER_BLOCK + wave * 32;

    // A fragments for the two 16-row tiles.
    // 32-bit A 16x4 layout: VGPR0 = K0 (lanes 0-15) / K2 (lanes 16-31),
    //                       VGPR1 = K1 / K3.  K = {x, y, z, 0-pad}.
    v2f a0, a1;
    a0.x = hi ? sz[r0 + lm]      : sx[r0 + lm];
    a0.y = hi ? 0.0f             : sy[r0 + lm];
    a1.x = hi ? sz[r0 + 16 + lm] : sx[r0 + 16 + lm];
    a1.y = hi ? 0.0f             : sy[r0 + 16 + lm];

    // Row squared-norms for this lane's C-fragment rows (M = v or v+8).
    float r2row0[8], r2row1[8];
#pragma unroll
    for (int v = 0; v < 8; ++v) {
        int M  = hi ? v + 8 : v;
        int g0 = r0 + M;
        int g1 = r0 + 16 + M;
        r2row0[v] = sx[g0]*sx[g0] + sy[g0]*sy[g0] + sz[g0]*sz[g0];
        r2row1[v] = sx[g1]*sx[g1] + sy[g1]*sy[g1] + sz[g1]*sz[g1];
    }

    // Register-resident sorted top-30 of SQUARED distances (ascending).
    float bd[TOPK];
    int   bj[TOPK];
#pragma unroll
    for (int k = 0; k < TOPK; ++k) { bd[k] = 3.0e38f; bj[k] = 0; }

    float* myTile = &tile[wave][0];

    for (int j0 = 0; j0 < N_NODES; j0 += 16) {
        // B fragment (4x16), symmetric layout to A.
        v2f bf;
        bf.x = hi ? sz[j0 + lm] : sx[j0 + lm];
        bf.y = hi ? 0.0f        : sy[j0 + lm];

        int gj = j0 + lm;
        float r2c = sx[gj]*sx[gj] + sy[gj]*sy[gj] + sz[gj]*sz[gj];

        v8f c0 = {}; v8f c1 = {};
        c0 = __builtin_amdgcn_wmma_f32_16x16x4_f32(false, a0, false, bf,
                                                   (short)0, c0, false, false);
        c1 = __builtin_amdgcn_wmma_f32_16x16x4_f32(false, a1, false, bf,
                                                   (short)0, c1, false, false);

        // Gram -> clamped squared distance, stage 32x16 tile in LDS (no sqrt!).
#pragma unroll
        for (int v = 0; v < 8; ++v) {
            int M = hi ? v + 8 : v;
            float q0 = fmaxf(r2row0[v] + r2c - 2.0f * c0[v], 0.0f);
            float q1 = fmaxf(r2row1[v] + r2c - 2.0f * c1[v], 0.0f);
            myTile[M * TILE_LD + lm]        = q0;
            myTile[(16 + M) * TILE_LD + lm] = q1;
        }
        asm volatile("s_wait_dscnt 0" ::: "memory");   // wave's DS stores visible

        // Lane 'lane' owns global row r0+lane: 4x float4 read-back + min4 gate.
#pragma unroll
        for (int c4 = 0; c4 < 16; c4 += 4) {
            float4 qv = *(const float4*)&myTile[lane * TILE_LD + c4];
            float mn = fminf(fminf(qv.x, qv.y), fminf(qv.z, qv.w));
            if (mn < bd[TOPK - 1]) {              // rare after saturation
                float qe[4] = { qv.x, qv.y, qv.z, qv.w };
#pragma unroll
                for (int e = 0; e < 4; ++e) {     // ascending j: stable ties
                    float q = qe[e];
                    int   j = j0 + c4 + e;
                    if (q < bd[TOPK - 1]) {
                        bd[TOPK - 1] = q; bj[TOPK - 1] = j;
#pragma unroll
                        for (int m = TOPK - 1; m > 0; --m) {
                            if (bd[m] < bd[m - 1]) {
                                float td = bd[m]; bd[m] = bd[m - 1]; bd[m - 1] = td;
                                int   tj = bj[m]; bj[m] = bj[m - 1]; bj[m - 1] = tj;
                            }
                        }
                    }
                }
            }
        }
    }

    // Emit D_neighbors / E_idx: sqrt only on the 30 survivors per row.
    const int i = r0 + lane;
    const size_t base = ((size_t)b * N_NODES + i) * TOPK;
#pragma unroll
    for (int k = 0; k < TOPK; ++k) {
        out_D[base + k] = sqrtf(bd[k] + EPS_D);   // bd already clamped >= 0
        out_E[base + k] = (float)bj[k];
    }

    // RBF + positional encodings (read back just-written values: same-wave
    // store->load same address is kept in order).
    const float kFreq[8] = { 1.0f, 0.31622776601683794f, 0.1f,
                             0.03162277660168379f, 0.01f,
                             0.0031622776601683794f, 0.001f,
                             0.00031622776601683794f };   // 10000^(-2m/16)
    for (int k = 0; k < TOPK; ++k) {
        float d = out_D[base + k];
        int   j = (int)out_E[base + k];

        float* gs = out_gs + (base + k) * 15;
#pragma unroll
        for (int m = 0; m < 15; ++m) {
            float mu = (20.0f / 14.0f) * (float)m;        // linspace(0,20,15)
            float t  = (d - mu) * (15.0f / 20.0f);        // sigma = 20/15
            gs[m] = __expf(-t * t);
        }

        float doff = (float)j - (float)i;
        float* pe = out_pos + (base + k) * 16;
#pragma unroll
        for (int m = 0; m < 8; ++m) {
            float a = doff * kFreq[m];
            pe[m]     = __cosf(a);
            pe[8 + m] = __sinf(a);
        }
    }
}

// ---------------- Kernel B: local frames (AD) + frame-projected neighbors ------
struct V3 { float x, y, z; };
__device__ inline V3 vsub(V3 a, V3 b) { return {a.x - b.x, a.y - b.y, a.z - b.z}; }
__device__ inline float vdot(V3 a, V3 b) { return a.x*b.x + a.y*b.y + a.z*b.z; }
__device__ inline V3 vcross(V3 a, V3 b) {
    return {a.y*b.z - a.z*b.y, a.z*b.x - a.x*b.z, a.x*b.y - a.y*b.x};
}
__device__ inline V3 vnorm(V3 a) {                 // x / max(|x|, 1e-12)
    float inv = 1.0f / fmaxf(sqrtf(vdot(a, a)), 1e-12f);
    return {a.x * inv, a.y * inv, a.z * inv};
}
__device__ inline float fclip(float v) { return fminf(fmaxf(v, -1.0f + 1e-6f), 1.0f - 1e-6f); }

__global__ void orient_kernel(const float* __restrict__ coords,
                              const float* __restrict__ out_E,
                              float* __restrict__ out_AD,   // (b,n,3)
                              float* __restrict__ out_O,    // (b,n,k,3)
                              int nb)
{
    int tid = blockIdx.x * blockDim.x + threadIdx.x;
    int total = nb * N_NODES;
    if (tid >= total) return;
    int b = tid / N_NODES;
    int i = tid % N_NODES;
    const float* Xb = coords + (size_t)b * N_NODES * 3;

    auto ld = [&](int n) -> V3 {
        return { Xb[n*3+0], Xb[n*3+1], Xb[n*3+2] };
    };

    V3 o1 = {0,0,0}, n2 = {0,0,0}, c3 = {0,0,0};
    float ad0 = 0.0f, ad1 = 0.0f, ad2 = 0.0f;

    const bool interior = (i >= 1) && (i <= N_NODES - 3);  // pad (1,2) zeros
    if (interior) {
        V3 xm1 = ld(i - 1), x0 = ld(i), xp1 = ld(i + 1), xp2 = ld(i + 2);
        V3 u2 = vnorm(vsub(x0,  xm1));   // U[i-1]
        V3 u1 = vnorm(vsub(xp1, x0));    // U[i]
        V3 u0 = vnorm(vsub(xp2, xp1));   // U[i+1]
        V3 n_2 = vnorm(vcross(u2, u1));
        V3 n_1 = vnorm(vcross(u1, u0));
        float A = acosf(fclip(-vdot(u1, u0)));
        float cD = fclip(vdot(n_2, n_1));
        float sgn = vdot(u2, n_1);
        float s = (sgn > 0.0f) ? 1.0f : ((sgn < 0.0f) ? -1.0f : 0.0f);
        float Dang = s * acosf(cD);
        ad0 = __cosf(A);
        ad1 = __sinf(A) * __cosf(Dang);
        ad2 = __sinf(A) * __sinf(Dang);
        o1 = vnorm(vsub(u2, u1));
        n2 = n_2;
        c3 = vcross(o1, n2);
    }

    size_t nbase = (size_t)b * N_NODES + i;
    out_AD[nbase * 3 + 0] = ad0;
    out_AD[nbase * 3 + 1] = ad1;
    out_AD[nbase * 3 + 2] = ad2;

    V3 xi = ld(i);
    for (int k = 0; k < TOPK; ++k) {
        size_t e = nbase * TOPK + k;
        int j = (int)out_E[e];
        V3 v = vsub(ld(j), xi);
        V3 du = vnorm(V3{ vdot(o1, v), vdot(n2, v), vdot(c3, v) });
        out_O[e * 3 + 0] = du.x;
        out_O[e * 3 + 1] = du.y;
        out_O[e * 3 + 2] = du.z;
    }
}

// ---------------- Host launcher -------------------------------------------------
extern "C" void kernel_launch(void* const* d_in, const int* in_sizes, int n_in,
                              void* d_out, int out_size, void* d_ws, size_t ws_size,
                              hipStream_t stream) {
    (void)n_in; (void)out_size; (void)d_ws; (void)ws_size;
    const float* coords = (const float*)d_in[0];
    const int nb = in_sizes[0] / (N_NODES * 3);   // batch count (=2)

    float* out = (float*)d_out;
    const size_t npos = (size_t)nb * N_NODES * TOPK * 16;
    const size_t nAD  = (size_t)nb * N_NODES * 3;
    const size_t nO   = (size_t)nb * N_NODES * TOPK * 3;
    const size_t ngs  = (size_t)nb * N_NODES * TOPK * 15;
    const size_t nD   = (size_t)nb * N_NODES * TOPK;

    float* out_pos = out;                 // 1. pos_emb
    float* out_AD  = out_pos + npos;      // 2. AD_features
    float* out_O   = out_AD + nAD;        // 3. O_features
    float* out_gs  = out_O + nO;          // 4. gs_d
    float* out_D   = out_gs + ngs;        // 5. D_neighbors
    float* out_E   = out_D + nD;          // 6. E_idx (as float)

    dim3 gridA(nb * (N_NODES / ROWS_PER_BLOCK));
    knn_feat_kernel<<<gridA, ROWS_PER_BLOCK, 0, stream>>>(coords, out_pos, out_gs,
                                                          out_D, out_E);

    int total = nb * N_NODES;
    orient_kernel<<<(total + 255) / 256, 256, 0, stream>>>(coords, out_E,
                                                           out_AD, out_O, nb);
}